// IWT_31628139168507
// MI455X (gfx1250) — compile-verified
//
#include <hip/hip_runtime.h>

// Inverse 2x2 Haar synthesis (IWT), MI455X / gfx1250, wave32.
//
// out[b,c,2h+p,2w+q] = sum_k x[b,k,c,h,w] * Wk[p,q],  Wk = +-1/4 butterfly.
//
// Pure HBM-bandwidth problem (1 GiB traffic, ~1 FLOP/byte -> ~46us floor at
// 23.3 TB/s). The 4-subband butterfly is a K=4 contraction, mapped onto
// V_WMMA_F32_16X16X4_F32: D[16,16] = A[16,4] x B[4,16], with A = constant
// butterfly (rows 0-3, rest zero) and B = 16 pixels' subband values.
// Each wave-iteration handles 64 consecutive-w pixels (4 WMMA groups) so the
// index decode + 64-bit address setup is amortized over 2 KiB of traffic.

typedef __attribute__((ext_vector_type(2))) float v2f;
typedef __attribute__((ext_vector_type(8))) float v8f;

namespace {
constexpr unsigned kC   = 64;            // channels per subband
constexpr unsigned kH   = 256;
constexpr unsigned kW   = 256;
constexpr unsigned kHW  = kH * kW;       // 2^16
constexpr unsigned kCHW = kC * kHW;      // 2^22  (subband plane stride, elements)
constexpr unsigned kB   = 8;
constexpr unsigned kPix = kB * kCHW;     // 2^25  input pixel positions
constexpr unsigned kQuads = kPix / 64;   // 2^19  64-pixel quads (one per wave-iter)
}

__global__ __launch_bounds__(256) void iwt_haar_wmma(const float* __restrict__ x,
                                                     float* __restrict__ out) {
    const unsigned lane = threadIdx.x & 31u;
    const unsigned n    = lane & 15u;    // column (pixel) index within a 16-group
    const bool     hi   = lane >= 16u;   // upper half-wave holds K=2,3 (HL,HH)

    // Constant A matrix (16x4), ISA layout: reg0 = A[m, hi?2:0], reg1 = A[m, hi?3:1].
    // Rows: m=0: ( .25, .25, .25, .25)   -> out(0,0)
    //       m=1: ( .25, .25,-.25,-.25)   -> out(0,1)
    //       m=2: ( .25,-.25, .25,-.25)   -> out(1,0)
    //       m=3: ( .25,-.25,-.25, .25)   -> out(1,1)   rows 4..15 = 0
    v2f a;
    {
        float ax = 0.0f, ay = 0.0f;
        if (n == 0u)      { ax = 0.25f;                 ay = 0.25f; }
        else if (n == 1u) { ax = hi ? -0.25f : 0.25f;   ay = hi ? -0.25f : 0.25f; }
        else if (n == 2u) { ax = 0.25f;                 ay = -0.25f; }
        else if (n == 3u) { ax = hi ? -0.25f : 0.25f;   ay = hi ?  0.25f : -0.25f; }
        a.x = ax; a.y = ay;
    }

    const unsigned waveId = (blockIdx.x * blockDim.x + threadIdx.x) >> 5;
    const unsigned nWaves = (gridDim.x * blockDim.x) >> 5;
    const unsigned laneInOff = (hi ? 2u * kCHW : 0u) + n;  // subband-half + pixel

    // Wave-uniform grid-stride loop => EXEC is all-1s at every WMMA.
    for (unsigned q = waveId; q < kQuads; q += nWaves) {
        const unsigned p0 = q << 6;               // first pixel of quad (flat b,c,h,w)
        const unsigned b  = p0 >> 22;             // / CHW
        const unsigned r  = p0 & (kCHW - 1u);     // c*HW + h*W + w0 (w0 multiple of 64)

        // B matrices for 4 groups of 16 pixels:
        //   lane n    -> B[0,n]=LL, B[1,n]=LH
        //   lane n+16 -> B[2,n]=HL, B[3,n]=HH
        const float* pin = x + (size_t)(b << 2) * kCHW + r + laneInOff;
        v2f bm[4];
#pragma unroll
        for (int j = 0; j < 4; ++j) {
            bm[j].x = __builtin_nontemporal_load(pin + 16 * j);         // K = hi?2:0
            bm[j].y = __builtin_nontemporal_load(pin + 16 * j + kCHW);  // K = hi?3:1
        }

        v8f acc[4];
#pragma unroll
        for (int j = 0; j < 4; ++j) {
            v8f z = {};
            acc[j] = __builtin_amdgcn_wmma_f32_16x16x4_f32(
                /*neg_a=*/false, a, /*neg_b=*/false, bm[j],
                /*c_mod=*/(short)0, z, /*reuse_a=*/false, /*reuse_b=*/false);
        }

        // D layout: VGPR r, lanes 0-15 = row M=r, N=lane.
        // acc[j][0..3] (lanes 0-15) = out00,out01,out10,out11 of pixel n of group j.
        if (!hi) {
            const unsigned c  = r >> 16;          // / HW
            const unsigned h  = (r >> 8) & 255u;  // / W % H
            const unsigned w0 = r & 255u;
            float* po = out + (((size_t)((b << 6) + c)) << 18)  // (b*C+c) * (2H*2W)
                            + ((size_t)h << 10)                 // 2h * 2W
                            + 2u * w0 + 2u * n;
#pragma unroll
            for (int j = 0; j < 4; ++j) {
                v2f o0; o0.x = acc[j][0]; o0.y = acc[j][1];     // row 2h
                v2f o1; o1.x = acc[j][2]; o1.y = acc[j][3];     // row 2h+1
                __builtin_nontemporal_store(o0, (v2f*)(po + 32 * j));
                __builtin_nontemporal_store(o1, (v2f*)(po + 32 * j + 2u * kW));
            }
        }
    }
}

extern "C" void kernel_launch(void* const* d_in, const int* in_sizes, int n_in,
                              void* d_out, int out_size, void* d_ws, size_t ws_size,
                              hipStream_t stream) {
    (void)in_sizes; (void)n_in; (void)out_size; (void)d_ws; (void)ws_size;
    const float* x = (const float*)d_in[0];
    float* out = (float*)d_out;
    // 4096 blocks x 256 threads = 32768 wave32s; 16 quads (2 KiB traffic each) per wave.
    iwt_haar_wmma<<<4096, 256, 0, stream>>>(x, out);
}